// WordSmoothCriterion_5755256177164
// MI455X (gfx1250) — compile-verified
//
#include <hip/hip_runtime.h>
#include <hip/hip_bf16.h>

typedef __attribute__((ext_vector_type(2))) float v2f;
typedef __attribute__((ext_vector_type(4))) float v4f;
typedef __attribute__((ext_vector_type(8))) float v8f;

#define MARGIN_C   0.5f
#define INV_TAU_C  (1.0f / 0.3f)
#define ALPHA_C    0.7f

__device__ __forceinline__ float smooth_elem(float s) {
    // sim = sim * (sim >= MARGIN)
    float cs = (s >= MARGIN_C) ? s : 0.0f;
    // exp((sim - 1)/tau) * (sim < 1)
    float sm = __expf((cs - 1.0f) * INV_TAU_C);
    return (cs < 1.0f) ? sm : 0.0f;
}

// One block per row. 8 wave32s stream the row in 128-element chunks (b128
// loads). The three running sums ride the matrix pipe via
// V_WMMA_F32_16X16X4_F32 with an all-ones B (row-sum trick; total = sum(D)/16)
// while VALU/TRANS do clip/exp. The <=127-element tail is folded in with
// scalar VALU so the hot loop has no guards.
__global__ __launch_bounds__(256) void wsc_row_kernel(
    const float* __restrict__ inp,    // [R, V] log-probs
    const float* __restrict__ simM,   // [V, V]
    const int*   __restrict__ tgt,    // [R]
    const float* __restrict__ mask,   // [R]
    float*       __restrict__ partial,// [R, 6]
    int V)
{
    const int r    = blockIdx.x;
    const int tid  = threadIdx.x;
    const int lane = tid & 31;
    const int wave = tid >> 5;

    const float* __restrict__ inp_row = inp + (size_t)r * (size_t)V;
    const int    t = tgt[r];
    const float* __restrict__ sim_row = simM + (size_t)t * (size_t)V;
    const float  m = mask[r];

    v8f c_s  = {};   // sum smooth
    v8f c_s2 = {};   // sum smooth^2
    v8f c_iw = {};   // sum inp*smooth
    const v2f ones = {1.0f, 1.0f};

    const int nFull = (V >> 7) << 7;            // elements covered by WMMA loop
    for (int idx = (wave << 7) + (lane << 2); idx < nFull; idx += 1024) {
        v4f sv = *(const v4f*)(sim_row + idx);                        // RT: re-used via L2
        v4f iv = __builtin_nontemporal_load((const v4f*)(inp_row + idx)); // NT: streamed once

        v2f sm0, sm1, sq0, sq1, iw0, iw1;
        sm0.x = smooth_elem(sv.x); sm0.y = smooth_elem(sv.y);
        sm1.x = smooth_elem(sv.z); sm1.y = smooth_elem(sv.w);
        sq0.x = sm0.x * sm0.x;     sq0.y = sm0.y * sm0.y;
        sq1.x = sm1.x * sm1.x;     sq1.y = sm1.y * sm1.y;
        iw0.x = iv.x * sm0.x;      iw0.y = iv.y * sm0.y;
        iw1.x = iv.z * sm1.x;      iw1.y = iv.w * sm1.y;

        // EXEC is all-ones here; D = A x ones + C accumulates row sums.
        c_s  = __builtin_amdgcn_wmma_f32_16x16x4_f32(false, sm0, false, ones, (short)0, c_s,  false, false);
        c_s2 = __builtin_amdgcn_wmma_f32_16x16x4_f32(false, sq0, false, ones, (short)0, c_s2, false, false);
        c_iw = __builtin_amdgcn_wmma_f32_16x16x4_f32(false, iw0, false, ones, (short)0, c_iw, false, false);
        c_s  = __builtin_amdgcn_wmma_f32_16x16x4_f32(false, sm1, false, ones, (short)0, c_s,  false, false);
        c_s2 = __builtin_amdgcn_wmma_f32_16x16x4_f32(false, sq1, false, ones, (short)0, c_s2, false, false);
        c_iw = __builtin_amdgcn_wmma_f32_16x16x4_f32(false, iw1, false, ones, (short)0, c_iw, false, false);
    }

    // Lane-local reduce across the 8 accumulator VGPRs; WMMA replicated each
    // row sum across 16 columns, so scale by 1/16 before mixing with the tail.
    const float inv16 = 1.0f / 16.0f;
    float ls = 0.0f, ls2 = 0.0f, liw = 0.0f;
#pragma unroll
    for (int k = 0; k < 8; ++k) { ls += c_s[k]; ls2 += c_s2[k]; liw += c_iw[k]; }
    ls *= inv16; ls2 *= inv16; liw *= inv16;

    // Tail (< 128 elements): plain VALU accumulation, one element per thread.
    const int tail = V - nFull;
    if (tid < tail) {
        float s  = sim_row[nFull + tid];
        float iv = inp_row[nFull + tid];
        float sm = smooth_elem(s);
        ls  += sm;
        ls2 += sm * sm;
        liw += iv * sm;
    }

    __shared__ float red0[256];
    __shared__ float red1[256];
    __shared__ float red2[256];
    red0[tid] = ls; red1[tid] = ls2; red2[tid] = liw;
    __syncthreads();
    for (int s = 128; s > 0; s >>= 1) {
        if (tid < s) {
            red0[tid] += red0[tid + s];
            red1[tid] += red1[tid + s];
            red2[tid] += red2[tid + s];
        }
        __syncthreads();
    }

    if (tid == 0) {
        float Ssm  = red0[0];
        float Ssm2 = red1[0];
        float Siw  = red2[0];
        float* p = partial + (size_t)r * 6;
        p[0] = Ssm;              // for word_mean
        p[1] = Ssm2;             // for word_std
        p[2] = m * Ssm;          // sum of w (mask * smooth)
        p[3] = m * Siw;          // sum of inp * w   (negated later)
        p[4] = m * inp_row[t];   // masked logp(target)
        p[5] = m;                // mask sum
    }
}

// Deterministic final reduction (single block, double accumulation).
__global__ __launch_bounds__(256) void wsc_final_kernel(
    const float* __restrict__ partial,  // [R, 6]
    float*       __restrict__ out,      // [4] = (ml, total, mean, std)
    int R, int V)
{
    __shared__ double red[6][256];
    const int tid = threadIdx.x;
    double a[6] = {0, 0, 0, 0, 0, 0};
    for (int r = tid; r < R; r += 256) {
        const float* p = partial + (size_t)r * 6;
#pragma unroll
        for (int k = 0; k < 6; ++k) a[k] += (double)p[k];
    }
#pragma unroll
    for (int k = 0; k < 6; ++k) red[k][tid] = a[k];
    __syncthreads();
    for (int s = 128; s > 0; s >>= 1) {
        if (tid < s) {
#pragma unroll
            for (int k = 0; k < 6; ++k) red[k][tid] += red[k][tid + s];
        }
        __syncthreads();
    }
    if (tid == 0) {
        double Ssm  = red[0][0];
        double Ssm2 = red[1][0];
        double Sw   = red[2][0];
        double Siw  = red[3][0];
        double Slp  = red[4][0];
        double Sm   = red[5][0];

        double N    = (double)R * (double)V;
        double mean = Ssm / N;
        double var  = Ssm2 / N - mean * mean;
        if (var < 0.0) var = 0.0;
        double stdv = sqrt(var);

        double ml     = -Slp / Sm;
        double output = -Siw / Sw;
        double total  = (double)ALPHA_C * output + (1.0 - (double)ALPHA_C) * ml;

        out[0] = (float)ml;
        out[1] = (float)total;
        out[2] = (float)mean;
        out[3] = (float)stdv;
    }
}

extern "C" void kernel_launch(void* const* d_in, const int* in_sizes, int n_in,
                              void* d_out, int out_size, void* d_ws, size_t ws_size,
                              hipStream_t stream) {
    const float* inp  = (const float*)d_in[0];   // [B,T,V] log-probs
    const float* simM = (const float*)d_in[1];   // [V,V]
    const int*   tgt  = (const int*)d_in[2];     // [B,T]
    const float* mask = (const float*)d_in[3];   // [B,T]

    const int R = in_sizes[2];            // B*T rows
    const int V = in_sizes[0] / R;        // vocab

    float* partial = (float*)d_ws;        // R * 6 floats

    wsc_row_kernel<<<R, 256, 0, stream>>>(inp, simM, tgt, mask, partial, V);
    wsc_final_kernel<<<1, 256, 0, stream>>>(partial, (float*)d_out, R, V);
}